// event_corr_block_88201448391152
// MI455X (gfx1250) — compile-verified
//
#include <hip/hip_runtime.h>

typedef float v2f __attribute__((ext_vector_type(2)));
typedef float v8f __attribute__((ext_vector_type(8)));

#if defined(__has_builtin)
#if __has_builtin(__builtin_amdgcn_wmma_f32_16x16x4_f32)
#define USE_WMMA_F32 1
#endif
#endif

namespace {
constexpr int Bn = 2, Cn = 128, Hn = 64, Wn = 96;
constexpr int HW = Hn * Wn;                // 6144
constexpr int OUT = 128;
constexpr int KOFF = 81, NL = 3;
constexpr int FEAT = 243, FPAD = 244, FSTR = 245;  // K padded to mult of 4; LDS stride odd vs 64 banks
constexpr int KSTEPS = FPAD / 4;           // 61 WMMA k-steps
constexpr int YSTR = 129;
constexpr int GROUPS = 8;
constexpr int TILES_PER_B = HW / 16;       // 384
constexpr int NTILES = Bn * TILES_PER_B;   // 768
constexpr float INV_SQRT_C = 0.08838834764831845f;  // 1/sqrt(128)

// workspace layout (in floats)
constexpr size_t SZ_T     = (size_t)Bn * HW * Cn;           // 1,572,864
constexpr size_t OFF_F1T  = 0;
constexpr size_t OFF_L0T  = OFF_F1T + SZ_T;
constexpr size_t SZ_L1    = (size_t)Bn * 32 * 48 * Cn;
constexpr size_t OFF_L1T  = OFF_L0T + SZ_T;
constexpr size_t SZ_L2    = (size_t)Bn * 16 * 24 * Cn;
constexpr size_t OFF_L2T  = OFF_L1T + SZ_L1;
constexpr size_t SZ_WP    = (size_t)GROUPS * KSTEPS * 32 * 2;  // 31,232 (fragment-packed B)
constexpr size_t OFF_WP   = OFF_L2T + SZ_L2;
constexpr size_t OFF_Y    = OFF_WP + SZ_WP;
constexpr size_t SZ_PART  = (size_t)Bn * GROUPS * TILES_PER_B * 2;
constexpr size_t OFF_PART = OFF_Y + SZ_T;
constexpr size_t OFF_MV   = OFF_PART + SZ_PART;
} // namespace

// ---------------------------------------------------------------------------
// Prep: NCHW -> NHWC transpose of fmap1/fmap2 (makes per-pixel 128-ch vectors
// contiguous so a wave32 reads them as 32 coalesced float4s).
__global__ __launch_bounds__(256) void prep_transpose(
    const float* __restrict__ f1, const float* __restrict__ f2,
    float* __restrict__ f1t, float* __restrict__ l0t) {
  const size_t idx = (size_t)blockIdx.x * 256 + threadIdx.x;
  if (idx >= SZ_T) return;
  const int c = (int)(idx & (Cn - 1));
  const int p = (int)((idx >> 7) % HW);
  const int b = (int)(idx / ((size_t)Cn * HW));
  const size_t src = ((size_t)b * Cn + c) * HW + p;
  f1t[idx] = f1[src];
  l0t[idx] = f2[src];
}

// 2x2 mean pool in NHWC (pooling fmap2 == pooling corr over target dims)
__global__ __launch_bounds__(256) void prep_pool(
    const float* __restrict__ src, float* __restrict__ dst, int Hs, int Ws, int total) {
  const int idx = blockIdx.x * 256 + threadIdx.x;
  if (idx >= total) return;
  const int Wd = Ws >> 1, Hd = Hs >> 1;
  const int c = idx & (Cn - 1);
  const int x = (idx >> 7) % Wd;
  const int y = (idx >> 7) / Wd % Hd;
  const int b = idx / (Cn * Wd * Hd);
  const float* s = src + (size_t)b * Hs * Ws * Cn;
  const size_t r0 = ((size_t)(2 * y) * Ws + 2 * x) * Cn + c;
  const size_t r1 = ((size_t)(2 * y + 1) * Ws + 2 * x) * Cn + c;
  dst[idx] = 0.25f * (s[r0] + s[r0 + Cn] + s[r1] + s[r1 + Cn]);
}

// conv_w [128,243] -> WMMA-fragment-packed B: wPack[g][t][lane] = float2 with
// (w(k,n), w(k+1,n)), k = 4t + 2*(lane>=16), n = 16g + (lane&15). One
// coalesced global_load_b64 per wave per WMMA k-step, zero address math.
__global__ __launch_bounds__(256) void prep_w(
    const float* __restrict__ cw, float* __restrict__ wPack) {
  const int idx = blockIdx.x * 256 + threadIdx.x;             // fragment pair id
  if (idx >= GROUPS * KSTEPS * 32) return;
  const int lane = idx & 31;
  const int t = (idx >> 5) % KSTEPS;
  const int g = idx / (KSTEPS * 32);
  const int k = 4 * t + 2 * (lane >> 4);
  const int n = g * 16 + (lane & 15);
  const float w0 = (k < FEAT) ? cw[(size_t)n * FEAT + k] : 0.f;
  const float w1 = (k + 1 < FEAT) ? cw[(size_t)n * FEAT + k + 1] : 0.f;
  wPack[(size_t)idx * 2 + 0] = w0;
  wPack[(size_t)idx * 2 + 1] = w1;
}

// ---------------------------------------------------------------------------
// Fused on-the-fly correlation sampling + 1x1 conv (WMMA) + GN tile partials.
// One workgroup = 8 waves = 16 pixels. Wave w samples pixels 2w, 2w+1:
// per pixel/level it builds a 10x10 grid of corner dot-products (dx,dy are
// integers so all 81 bilinear samples share one fractional weight pair),
// then blends rows on the fly into the LDS feature tile.
__global__ __launch_bounds__(256) void corr_sample_conv(
    const float* __restrict__ f1t, const float* __restrict__ l0t,
    const float* __restrict__ l1t, const float* __restrict__ l2t,
    const float* __restrict__ txy, const float* __restrict__ wPack,
    const float* __restrict__ bias, float* __restrict__ ybuf,
    float* __restrict__ partial) {
  __shared__ float feat[16 * FSTR];
  __shared__ float ytile[16 * YSTR];

  const int tid  = threadIdx.x;
  const int wave = tid >> 5;
  const int lane = tid & 31;
  const int tile = blockIdx.x;                 // 0..767
  const int b    = tile / TILES_PER_B;
  const int tb   = tile - b * TILES_PER_B;
  const int pb0  = tb * 16;                    // base pixel within batch

  // ---------------- sampling phase ----------------
  for (int mi = 0; mi < 2; ++mi) {
    const int m  = wave * 2 + mi;
    const int pb = pb0 + m;
    // lane's 4 channels of the fmap1 pixel vector (coalesced 512B per wave)
    const float4 f1 = *((const float4*)(f1t + ((size_t)b * HW + pb) * Cn) + lane);
    const float tx = txy[((size_t)b * 2 + 0) * HW + pb];
    const float ty = txy[((size_t)b * 2 + 1) * HW + pb];
    if (lane == 0) feat[m * FSTR + FEAT] = 0.f;  // K-pad column

    for (int l = 0; l < NL; ++l) {
      const float inv = (l == 0) ? 1.f : ((l == 1) ? 0.5f : 0.25f);
      const int Hl = Hn >> l, Wl = Wn >> l;
      const float* lvl = (l == 0) ? l0t : ((l == 1) ? l1t : l2t);
      const float* lb = lvl + (size_t)b * Hl * Wl * Cn;
      const float xc = tx * inv, yc = ty * inv;
      const float xf = floorf(xc), yf = floorf(yc);
      const int x0 = (int)xf - 4, y0 = (int)yf - 4;
      const float wx1 = xc - xf, wx0 = 1.f - wx1;
      const float wy1 = yc - yf, wy0 = 1.f - wy1;

      float prev[10], cur[10];
      for (int r = 0; r < 10; ++r) {
        const int yi = y0 + r;
        const bool yok = (yi >= 0) && (yi < Hl);
        const int ycl = yi < 0 ? 0 : (yi > Hl - 1 ? Hl - 1 : yi);
#pragma unroll
        for (int c = 0; c < 10; ++c) {
          const int xi = x0 + c;
          const bool ok = yok && (xi >= 0) && (xi < Wl);
          const int xcl = xi < 0 ? 0 : (xi > Wl - 1 ? Wl - 1 : xi);
          const float4 v = *((const float4*)(lb + ((size_t)ycl * Wl + xcl) * Cn) + lane);
          float part = f1.x * v.x + f1.y * v.y + f1.z * v.z + f1.w * v.w;
#pragma unroll
          for (int off = 16; off; off >>= 1) part += __shfl_xor(part, off, 32);
          cur[c] = ok ? part * INV_SQRT_C : 0.f;
        }
        if (r > 0 && lane == 0) {
#pragma unroll
          for (int dxi = 0; dxi < 9; ++dxi) {
            const float val = wy0 * (wx0 * prev[dxi] + wx1 * prev[dxi + 1]) +
                              wy1 * (wx0 * cur[dxi]  + wx1 * cur[dxi + 1]);
            feat[m * FSTR + l * KOFF + (r - 1) * 9 + dxi] = val;
          }
        }
#pragma unroll
        for (int c = 0; c < 10; ++c) prev[c] = cur[c];
      }
    }
  }
  __syncthreads();

  // ---------------- 1x1 conv: 16x128 = 8 WMMA n-tiles, one per wave --------
  const int g  = wave;
  const int lo = lane & 15;
  const int hi = lane >> 4;
  const int n  = g * 16 + lo;
#if defined(USE_WMMA_F32)
  {
    const v2f* wp = (const v2f*)wPack + (size_t)g * KSTEPS * 32 + lane;
    v8f acc = {};
    for (int t = 0; t < KSTEPS; ++t) {
      const int ka = 4 * t + 2 * hi;  // A: lanes<16 hold K=k0,k0+1; lanes>=16 K=k0+2,k0+3
      v2f a;
      a[0] = feat[lo * FSTR + ka];
      a[1] = feat[lo * FSTR + ka + 1];
      const v2f bm = wp[t * 32];       // one global_load_b64, coalesced per wave
      acc = __builtin_amdgcn_wmma_f32_16x16x4_f32(
          false, a, false, bm, (short)0, acc, false, false);
    }
    const float bn = bias[n];
#pragma unroll
    for (int r = 0; r < 8; ++r)                  // C/D layout: M = r + 8*hi, N = n
      ytile[(r + 8 * hi) * YSTR + n] = acc[r] + bn;
  }
#else
  for (int idx = tid; idx < 16 * OUT; idx += 256) {
    const int mm = idx & 15, nn = idx >> 4;
    const int gg = nn >> 4, ll = nn & 15;
    float a2 = bias[nn];
    for (int t = 0; t < KSTEPS; ++t)
      for (int pos = 0; pos < 4; ++pos) {
        const int k = 4 * t + pos;
        if (k < FEAT)
          a2 += feat[mm * FSTR + k] *
                wPack[(((size_t)gg * KSTEPS + t) * 32 + (pos >> 1) * 16 + ll) * 2 + (pos & 1)];
      }
    ytile[mm * YSTR + nn] = a2;
  }
#endif
  __syncthreads();

  // ---------------- epilogue: store pre-GN y + deterministic GN partials ---
  // m-fast mapping: each 16-lane half-wave writes 64B contiguous in ybuf;
  // LDS reads hit distinct banks (stride 129 = 1 mod 64).
  for (int idx = tid; idx < 16 * OUT; idx += 256) {
    const int m = idx & 15;
    const int nn = idx >> 4;
    ybuf[((size_t)b * OUT + nn) * HW + pb0 + m] = ytile[m * YSTR + nn];
  }
  {
    float s = 0.f, ss = 0.f;                     // group g == n-tile g (width 16)
#pragma unroll
    for (int r = 0; r < 8; ++r) {
      const float v = ytile[(r + 8 * hi) * YSTR + g * 16 + lo];
      s += v; ss += v * v;
    }
#pragma unroll
    for (int off = 16; off; off >>= 1) {
      s += __shfl_xor(s, off, 32);
      ss += __shfl_xor(ss, off, 32);
    }
    if (lane == 0) {
      const size_t pi = (((size_t)b * GROUPS + g) * TILES_PER_B + tb) * 2;
      partial[pi] = s;
      partial[pi + 1] = ss;
    }
  }
}

// ---------------------------------------------------------------------------
// Deterministic tree reduction of the 384 tile partials per (batch, group).
__global__ __launch_bounds__(256) void gn_reduce(
    const float* __restrict__ partial, float* __restrict__ mv) {
  __shared__ float shs[256], shq[256];
  const int bg = blockIdx.x;                     // 0..15
  const int t = threadIdx.x;
  float s = 0.f, ss = 0.f;
  for (int i = t; i < TILES_PER_B; i += 256) {
    s  += partial[((size_t)bg * TILES_PER_B + i) * 2 + 0];
    ss += partial[((size_t)bg * TILES_PER_B + i) * 2 + 1];
  }
  shs[t] = s; shq[t] = ss;
  __syncthreads();
  for (int o = 128; o; o >>= 1) {
    if (t < o) { shs[t] += shs[t + o]; shq[t] += shq[t + o]; }
    __syncthreads();
  }
  if (t == 0) {
    const float N = 16.f * (float)HW;            // 98304 elems per (b,g)
    const float mean = shs[0] / N;
    const float var = shq[0] / N - mean * mean;
    mv[bg * 2 + 0] = mean;
    mv[bg * 2 + 1] = rsqrtf(var + 1e-5f);
  }
}

// GroupNorm affine + LeakyReLU(0.1)
__global__ __launch_bounds__(256) void gn_apply(
    const float* __restrict__ y, const float* __restrict__ mv,
    const float* __restrict__ gnw, const float* __restrict__ gnb,
    float* __restrict__ out) {
  const size_t idx = (size_t)blockIdx.x * 256 + threadIdx.x;
  if (idx >= SZ_T) return;
  const int n = (int)((idx / HW) & (OUT - 1));
  const int b = (int)(idx / ((size_t)OUT * HW));
  const int g = n >> 4;
  const float mean = mv[(b * GROUPS + g) * 2 + 0];
  const float istd = mv[(b * GROUPS + g) * 2 + 1];
  const float v = (y[idx] - mean) * istd * gnw[n] + gnb[n];
  out[idx] = v > 0.f ? v : 0.1f * v;
}

// ---------------------------------------------------------------------------
extern "C" void kernel_launch(void* const* d_in, const int* in_sizes, int n_in,
                              void* d_out, int out_size, void* d_ws, size_t ws_size,
                              hipStream_t stream) {
  (void)in_sizes; (void)n_in; (void)out_size; (void)ws_size;
  const float* fmap1  = (const float*)d_in[0];
  const float* fmap2  = (const float*)d_in[1];
  const float* txy    = (const float*)d_in[2];
  const float* conv_w = (const float*)d_in[3];
  const float* conv_b = (const float*)d_in[4];
  const float* gn_w   = (const float*)d_in[5];
  const float* gn_b   = (const float*)d_in[6];
  float* out = (float*)d_out;
  float* ws  = (float*)d_ws;

  float* f1t   = ws + OFF_F1T;
  float* l0t   = ws + OFF_L0T;
  float* l1t   = ws + OFF_L1T;
  float* l2t   = ws + OFF_L2T;
  float* wPack = ws + OFF_WP;
  float* ybuf  = ws + OFF_Y;
  float* part  = ws + OFF_PART;
  float* mv    = ws + OFF_MV;

  prep_transpose<<<(int)((SZ_T + 255) / 256), 256, 0, stream>>>(fmap1, fmap2, f1t, l0t);
  prep_pool<<<(int)((SZ_L1 + 255) / 256), 256, 0, stream>>>(l0t, l1t, Hn, Wn, (int)SZ_L1);
  prep_pool<<<(int)((SZ_L2 + 255) / 256), 256, 0, stream>>>(l1t, l2t, 32, 48, (int)SZ_L2);
  prep_w<<<(GROUPS * KSTEPS * 32 + 255) / 256, 256, 0, stream>>>(conv_w, wPack);

  corr_sample_conv<<<NTILES, 256, 0, stream>>>(f1t, l0t, l1t, l2t, txy, wPack,
                                               conv_b, ybuf, part);

  gn_reduce<<<Bn * GROUPS, 256, 0, stream>>>(part, mv);
  gn_apply<<<(int)((SZ_T + 255) / 256), 256, 0, stream>>>(ybuf, mv, gn_w, gn_b, out);
}